// LossWrap_77343771066384
// MI455X (gfx1250) — compile-verified
//
#include <hip/hip_runtime.h>
#include <hip/hip_bf16.h>

#define ROWF       152          // floats per prediction row (16 + 136)
#define YROWF      24
#define TILE_ROWS  32           // one row per lane, one tile per wave
#define BLOCK      32           // one wave per block (double-buffered LDS fits 64KB)
#define F4_PER_LANE ((TILE_ROWS * ROWF) / (4 * 32))   // 38 float4 async loads / lane / tile

typedef float v2f_t  __attribute__((ext_vector_type(2)));
typedef float v8f_t  __attribute__((ext_vector_type(8)));
typedef float f32x4  __attribute__((ext_vector_type(4)));  // builtin-compatible float4
typedef int   i32x4_t __attribute__((vector_size(16)));    // matches builtin's V4i

// ---------------- CDNA5-specific paths (probe-guarded) ----------------
#if defined(__gfx1250__) && __has_builtin(__builtin_amdgcn_global_load_async_to_lds_b128) && __has_builtin(__builtin_amdgcn_s_wait_asynccnt)
#define USE_ASYNC 1
#else
#define USE_ASYNC 0
#endif

__device__ __forceinline__ void stage_f4(const float* gsrc, float* ldst) {
#if USE_ASYNC
    __builtin_amdgcn_global_load_async_to_lds_b128(
        (__attribute__((address_space(1))) i32x4_t*)(void*)gsrc,
        (__attribute__((address_space(3))) i32x4_t*)(void*)ldst,
        /*imm offset*/0, /*cpol*/0);
#else
    *(f32x4*)ldst = *(const f32x4*)gsrc;
#endif
}

template <int N>
__device__ __forceinline__ void stage_wait_le() {
#if USE_ASYNC
    __builtin_amdgcn_s_wait_asynccnt(N);   // ASYNCcnt <= N; loads retire in order
#endif
    asm volatile("" ::: "memory");
}

// D(16x16) += ones(16x4) x B(4x16): column-sums 4 lanes' values per call.
__device__ __forceinline__ v8f_t wave_colsum4(v2f_t b, v8f_t c) {
#if defined(__gfx1250__) && __has_builtin(__builtin_amdgcn_wmma_f32_16x16x4_f32)
    v2f_t ones; ones[0] = 1.0f; ones[1] = 1.0f;
    return __builtin_amdgcn_wmma_f32_16x16x4_f32(
        false, ones, false, b, (short)0, c, false, false);
#else
    c[0] += b[0] + b[1];   // host-parse / fallback placeholder
    return c;
#endif
}

// ---------------- kernels ----------------
__global__ void ws_init_kernel(float* ws) {
    int t = threadIdx.x;
    if (t < 18) ws[t] = 0.0f;
}

__global__ __launch_bounds__(BLOCK)
void nll_main_kernel(const float* __restrict__ pred,
                     const float* __restrict__ y,
                     float* __restrict__ ws,
                     int tiles) {
    // two tile buffers per wave: 2 * 32 * 152 * 4 = 38912 B
    __shared__ __align__(16) float smem[2][TILE_ROWS * ROWF];

    const int lane = threadIdx.x & 31;
    const int gw   = (int)blockIdx.x;        // one wave per block
    const int nw   = (int)gridDim.x;

    float accE[16];
#pragma unroll
    for (int i = 0; i < 16; ++i) accE[i] = 0.0f;
    float accN = 0.0f, accL = 0.0f;

    auto issue_tile = [&](int t, float* buf) {
        const float* gbase = pred + (size_t)t * (TILE_ROWS * ROWF);
#pragma unroll
        for (int j = 0; j < F4_PER_LANE; ++j) {
            const int g4 = j * 32 + lane;                 // flat float4 index
            stage_f4(gbase + (size_t)g4 * 4, buf + g4 * 4);
        }
    };

    int t   = gw;
    int cur = 0;
    if (t < tiles) issue_tile(t, smem[0]);

    while (t < tiles) {
        const int  tn = t + nw;
        const bool pf = (tn < tiles);
        if (pf) issue_tile(tn, smem[cur ^ 1]);            // prefetch next tile
        if (pf) stage_wait_le<F4_PER_LANE>();             // older batch of 38 landed
        else    stage_wait_le<0>();

        // ---- per-lane row compute out of LDS ----
        const float* r   = smem[cur] + lane * ROWF;
        const size_t row = (size_t)t * TILE_ROWS + lane;
        const f32x4* pv  = (const f32x4*)r;
        const f32x4* yv  = (const f32x4*)(y + row * YROWF);
        f32x4 a0 = pv[0], a1 = pv[1], a2 = pv[2], a3 = pv[3];
        f32x4 b0 = __builtin_nontemporal_load(yv + 0);
        f32x4 b1 = __builtin_nontemporal_load(yv + 1);
        f32x4 b2 = __builtin_nontemporal_load(yv + 2);
        f32x4 b3 = __builtin_nontemporal_load(yv + 3);

        float res[16];
        res[0]  = a0.x - b0.x;  res[1]  = a0.y - b0.y;
        res[2]  = a0.z - b0.z;  res[3]  = a0.w - b0.w;
        res[4]  = a1.x - b1.x;  res[5]  = a1.y - b1.y;
        res[6]  = a1.z - b1.z;  res[7]  = a1.w - b1.w;
        res[8]  = a2.x - b2.x;  res[9]  = a2.y - b2.y;
        res[10] = a2.z - b2.z;  res[11] = a2.w - b2.w;
        res[12] = a3.x - b3.x;  res[13] = a3.y - b3.y;
        res[14] = a3.z - b3.z;  res[15] = a3.w - b3.w;

#pragma unroll
        for (int k = 0; k < 16; ++k) accE[k] = fmaf(res[k], res[k], accE[k]);

        // upper-triangular matvec + logdet (compile-time LDS offsets)
        const float* x = r + 16;
#pragma unroll
        for (int j = 0; j < 16; ++j) {
            float s = 0.0f;
#pragma unroll
            for (int k = j; k < 16; ++k) {
                const int xi = (j <= 7) ? (16 * j + k) : (271 - 16 * j - k);
                s = fmaf(x[xi], res[k], s);
            }
            accN = fmaf(s, s, accN);
            const float dv = (j <= 7) ? x[17 * j] : x[271 - 17 * j];
            const float dd = dv + 2.0e-30f;               // (diag+eps)+eps
            accL += logf(dd * dd);
        }

        cur ^= 1;
        t = tn;
    }

    // ---- wave-level reduction via V_WMMA_F32_16X16X4_F32 ----
    // tile buffers are dead now; reuse as 32-comp x 33-padded scratch
    __syncthreads();
    float* wbuf = smem[0];
#pragma unroll
    for (int c = 0; c < 16; ++c) wbuf[c * 33 + lane] = accE[c];
    wbuf[16 * 33 + lane] = accN;
    wbuf[17 * 33 + lane] = accL;
#pragma unroll
    for (int c = 18; c < 32; ++c) wbuf[c * 33 + lane] = 0.0f;
    __syncthreads();

    const int col  = lane & 15;
    const int half = lane >> 4;
    v8f_t d0, d1;
#pragma unroll
    for (int i = 0; i < 8; ++i) { d0[i] = 0.0f; d1[i] = 0.0f; }
#pragma unroll
    for (int g2 = 0; g2 < 8; ++g2) {
        v2f_t bb0, bb1;
        bb0[0] = wbuf[col * 33 + 4 * g2 + half];
        bb0[1] = wbuf[col * 33 + 4 * g2 + 2 + half];
        bb1[0] = wbuf[(16 + col) * 33 + 4 * g2 + half];
        bb1[1] = wbuf[(16 + col) * 33 + 4 * g2 + 2 + half];
        d0 = wave_colsum4(bb0, d0);   // comps 0..15
        d1 = wave_colsum4(bb1, d1);   // comps 16..31 (18..31 are zero)
    }
    // D column n lives in lanes n / n+16; VGPR0 holds row m=0 -> the sum
    if (lane < 16) atomicAdd(&ws[col], d0[0]);
    if (lane < 2)  atomicAdd(&ws[16 + col], d1[0]);
}

__global__ void finalize_kernel(const float* __restrict__ ws,
                                float* __restrict__ out, float invB) {
    int t = threadIdx.x;
    if (t < 16) out[3 + t] = ws[t] * invB;           // err[16]
    if (t == 0) {
        float mean_norm    =  ws[16] * invB;
        float mean_log_det = -ws[17] * invB;
        out[0] = mean_norm + mean_log_det;           // loss_total
        out[1] = mean_norm;
        out[2] = mean_log_det;
    }
}

// ---------------- launcher ----------------
extern "C" void kernel_launch(void* const* d_in, const int* in_sizes, int n_in,
                              void* d_out, int out_size, void* d_ws, size_t ws_size,
                              hipStream_t stream) {
    const float* pred = (const float*)d_in[0];   // [B, 152] f32
    const float* y    = (const float*)d_in[1];   // [B, 24]  f32
    float* out = (float*)d_out;                  // 19 f32
    float* ws  = (float*)d_ws;                   // 18 f32 partial sums

    const int batch = in_sizes[0] / ROWF;
    const int tiles = batch / TILE_ROWS;

    ws_init_kernel<<<1, 32, 0, stream>>>(ws);

    int blocks = 4096;                           // 4096 waves -> 8 tiles/wave at B=2^20
    if (blocks > tiles) blocks = tiles;
    if (blocks < 1) blocks = 1;
    nll_main_kernel<<<blocks, BLOCK, 0, stream>>>(pred, y, ws, tiles);

    finalize_kernel<<<1, 32, 0, stream>>>(ws, out, 1.0f / (float)batch);
}